// trajectory_predict_tensor_82772609728551
// MI455X (gfx1250) — compile-verified
//
#include <hip/hip_runtime.h>
#include <cstdint>
#include <math.h>

// Problem constants (from reference)
#define BATCH      262144
#define TDIM       80
#define STEPS      30
#define DT_F       0.1f

// Tiling
#define BLOCK      128          // threads per block == batch rows per block
#define COLS       32           // staged columns 48..79 (covers 50..79 needed)
#define COL0       48
#define LDS_STRIDE 33           // odd -> conflict-free column reads
#define OUT_STRIDE 31           // odd -> conflict-free column writes

typedef __attribute__((address_space(1))) int gint;   // global int
typedef __attribute__((address_space(3))) int lint;   // LDS int

// Integer round-trip casts: avoid addrspace-cast diagnostics.
// AS3 pointers are 32-bit LDS offsets (generic LDS addr low 32 bits).
__device__ __forceinline__ gint* as_global(const void* p) {
    return (gint*)(unsigned long long)(uintptr_t)p;
}
__device__ __forceinline__ lint* as_lds(const void* p) {
    return (lint*)(unsigned int)(uintptr_t)p;
}

__global__ __launch_bounds__(BLOCK) void idm_scan_kernel(
    const float* __restrict__ params,
    const float* __restrict__ sv_v1,
    const float* __restrict__ h1,
    const float* __restrict__ dv1,
    const float* __restrict__ sv_Y1,
    const float* __restrict__ lv_Y_seq,
    const float* __restrict__ lv_v_seq,
    float* __restrict__ out)
{
    __shared__ float tileY[BLOCK * LDS_STRIDE];
    __shared__ float tileV[BLOCK * LDS_STRIDE];
    __shared__ float outT [BLOCK * OUT_STRIDE];

    const int tid = threadIdx.x;
    const int b0  = blockIdx.x * BLOCK;

    // ---- Phase 1: async-stage lv_Y / lv_v tile (cols 48..79, 128 rows) ----
    // Mapping: per pass, 4 rows x 32 cols; wave32 = one row = 128B contiguous.
    {
        const int r0 = tid >> 5;        // 0..3
        const int c  = tid & 31;        // 0..31
        const float* gY = lv_Y_seq + (size_t)(b0 + r0) * TDIM + COL0 + c;
        const float* gV = lv_v_seq + (size_t)(b0 + r0) * TDIM + COL0 + c;
        const float* lY = &tileY[r0 * LDS_STRIDE + c];
        const float* lV = &tileV[r0 * LDS_STRIDE + c];
#pragma unroll
        for (int p = 0; p < 32; ++p) {
            __builtin_amdgcn_global_load_async_to_lds_b32(as_global(gY), as_lds(lY), 0, 0);
            __builtin_amdgcn_global_load_async_to_lds_b32(as_global(gV), as_lds(lV), 0, 0);
            gY += 4 * TDIM;             // advance 4 rows in global
            gV += 4 * TDIM;
            lY += 4 * LDS_STRIDE;       // advance 4 rows in LDS
            lV += 4 * LDS_STRIDE;
        }
    }

    // ---- Overlap: per-thread initial-state + params loads while DMA runs ----
    const size_t sbase = (size_t)(b0 + tid) * TDIM + 49;
    float v    = sv_v1[sbase];
    float gap  = h1[sbase];
    float dvv  = dv1[sbase];
    float y    = sv_Y1[sbase];

    const float s0  = params[0];
    const float Thw = params[1];
    const float pa  = params[2];
    const float pb  = params[3];
    const float v0  = params[4];
    const float inv2sab = 1.0f / (2.0f * sqrtf(pa * pb));
    const float inv_v0  = 1.0f / v0;

    __builtin_amdgcn_s_wait_asynccnt(0);   // this wave's DMA done
    __syncthreads();                       // all waves' tiles visible

    // ---- Phase 2: 30-step IDM scan, fully unrolled in registers ----
    const float* ty = &tileY[tid * LDS_STRIDE + (50 - COL0)];
    const float* tv = &tileV[tid * LDS_STRIDE + (50 - COL0)];
    float*       to = &outT [tid * OUT_STRIDE];

#pragma unroll
    for (int i = 0; i < STEPS; ++i) {
        float s_star = s0 + fmaxf(v * Thw + v * dvv * inv2sab, 0.0f);
        float q  = v * inv_v0;  q = q * q;  q = q * q;          // (v/v0)^4
        float rr = s_star / gap;                                 // (s*/h)
        float a_calc = pa * (1.0f - q - rr * rr);
        float v2  = v + a_calc * DT_F;
        float a_t = (v2 <= 0.0f) ? (-v / DT_F) : a_calc;
        float v_next = v + a_t * DT_F;

        float lvY = ty[i];
        float lvv = tv[i];
        dvv = v_next - lvv;
        y   = y + v_next * DT_F;
        gap = lvY - y;
        v   = v_next;
        to[i] = y;                      // stage output row in LDS
    }

    __syncthreads();                    // out-tile complete for all waves

    // ---- Phase 3: async coalesced store of (BLOCK x 30) output region ----
    {
        float* gout = out + (size_t)b0 * STEPS;   // 3840 contiguous floats
#pragma unroll
        for (int p = 0; p < STEPS; ++p) {
            int flat = p * BLOCK + tid;
            int r = flat / STEPS;
            int c = flat - r * STEPS;
            __builtin_amdgcn_global_store_async_from_lds_b32(
                as_global(gout + flat), as_lds(&outT[r * OUT_STRIDE + c]), 0, 0);
        }
    }
    __builtin_amdgcn_s_wait_asynccnt(0);
}

extern "C" void kernel_launch(void* const* d_in, const int* in_sizes, int n_in,
                              void* d_out, int out_size, void* d_ws, size_t ws_size,
                              hipStream_t stream) {
    (void)in_sizes; (void)n_in; (void)out_size; (void)d_ws; (void)ws_size;
    const float* params = (const float*)d_in[0];
    const float* sv_v1  = (const float*)d_in[1];
    const float* h1     = (const float*)d_in[2];
    const float* dv1    = (const float*)d_in[3];
    const float* sv_Y1  = (const float*)d_in[4];
    const float* lv_Y   = (const float*)d_in[5];
    const float* lv_v   = (const float*)d_in[6];
    float* out = (float*)d_out;

    dim3 grid(BATCH / BLOCK), block(BLOCK);
    hipLaunchKernelGGL(idm_scan_kernel, grid, block, 0, stream,
                       params, sv_v1, h1, dv1, sv_Y1, lv_Y, lv_v, out);
}